// ImageEvaluator_84000970375834
// MI455X (gfx1250) — compile-verified
//
#include <hip/hip_runtime.h>

// CDNA5 / gfx1250 wave32 WMMA types
typedef __attribute__((ext_vector_type(2))) float v2f;
typedef __attribute__((ext_vector_type(8))) float v8f;

#define H_IMG 512
#define W_IMG 512
#define CH_STRIDE (512 * 512)
#define RS 42   // LDS row stride (floats) for the 40-wide region (bank-conflict-free, 8B aligned)
#define HS 17   // LDS row stride for the 16-wide Hsum tiles

__global__ __launch_bounds__(32)
void ImageEvaluator_84000970375834_kernel(const float* __restrict__ img,
                                          float* __restrict__ out) {
    __shared__ float sXs[48 * RS];   // sum over channels of x     (rows 0..39 valid)
    __shared__ float sXq[48 * RS];   // sum over channels of x*x
    __shared__ float sH1[40 * HS];   // horizontal 25-tap box sum of sXs
    __shared__ float sH2[40 * HS];   // horizontal 25-tap box sum of sXq

    const int lane = threadIdx.x;          // 0..31 (wave32)
    const int hi   = lane >> 4;            // 0 or 1 (lane half)
    const int lo   = lane & 15;
    const int tc   = blockIdx.x * 16;      // tile origin col
    const int tr   = blockIdx.y * 16;      // tile origin row
    const int b    = blockIdx.z;

    const float* base = img + (size_t)b * 3 * CH_STRIDE;

    // ---- Band matrix chunks, shared by BOTH GEMM passes ----
    // band(j) = 1 iff (j - lo) in [0, 24]  -> single unsigned compare per element.
    // Chunk k holds logical K indices {4k+2*hi, 4k+2*hi+1} per the f32 A/B VGPR layout.
    v2f band[10];
    #pragma unroll
    for (int k = 0; k < 10; ++k) {
        const int j0 = 4 * k + 2 * hi;
        band[k].x = ((unsigned)(j0     - lo) <= 24u) ? 1.0f : 0.0f;
        band[k].y = ((unsigned)(j0 + 1 - lo) <= 24u) ? 1.0f : 0.0f;
    }

    // ---- Stage 1: load 40x40 reflect-padded region, fuse channel reduction ----
    for (int p = lane; p < 40 * 40; p += 32) {
        int rr = p / 40, cc = p - rr * 40;
        int ir = tr - 12 + rr; ir = (ir < 0) ? -ir : ((ir > 511) ? 1022 - ir : ir);
        int ic = tc - 12 + cc; ic = (ic < 0) ? -ic : ((ic > 511) ? 1022 - ic : ic);
        size_t off = (size_t)ir * W_IMG + ic;
        float x0 = base[off];
        float x1 = base[off + CH_STRIDE];
        float x2 = base[off + 2 * CH_STRIDE];
        sXs[rr * RS + cc] = x0 + x1 + x2;
        sXq[rr * RS + cc] = x0 * x0 + x1 * x1 + x2 * x2;
    }
    __syncthreads();

    // ---- Stage 2: horizontal banded GEMM via V_WMMA_F32_16X16X4_F32 ----
    // Hsum[r][n] = sum_{j=0..39} X[r][j] * S[j][n],  S[j][n] = band(j - n)
    #pragma unroll
    for (int g = 0; g < 3; ++g) {
        v8f acc1 = {}, acc2 = {};
        const int arow = g * 16 + lo;                 // A-matrix row held by this lane
        #pragma unroll
        for (int k = 0; k < 10; ++k) {
            const int j0 = 4 * k + 2 * hi;            // logical K indices j0, j0+1
            v2f a1 = { sXs[arow * RS + j0], sXs[arow * RS + j0 + 1] };
            v2f a2 = { sXq[arow * RS + j0], sXq[arow * RS + j0 + 1] };
            acc1 = __builtin_amdgcn_wmma_f32_16x16x4_f32(false, a1, false, band[k],
                                                         (short)0, acc1, false, false);
            acc2 = __builtin_amdgcn_wmma_f32_16x16x4_f32(false, a2, false, band[k],
                                                         (short)0, acc2, false, false);
        }
        // D layout: VGPR v -> row M = v + 8*hi, col N = lo
        #pragma unroll
        for (int v = 0; v < 8; ++v) {
            int row = g * 16 + v + 8 * hi;
            if (row < 40) {
                sH1[row * HS + lo] = acc1[v];
                sH2[row * HS + lo] = acc2[v];
            }
        }
    }
    __syncthreads();

    // ---- Stage 3: vertical banded GEMM (A is the same band matrix) ----
    // Out[m][n] = sum_{i=0..39} band(i - m) * Hsum[i][n]
    v8f s1 = {}, s2 = {};
    #pragma unroll
    for (int k = 0; k < 10; ++k) {
        const int i0 = 4 * k + 2 * hi;
        v2f b1 = { sH1[i0 * HS + lo], sH1[(i0 + 1) * HS + lo] };
        v2f b2 = { sH2[i0 * HS + lo], sH2[(i0 + 1) * HS + lo] };
        s1 = __builtin_amdgcn_wmma_f32_16x16x4_f32(false, band[k], false, b1,
                                                   (short)0, s1, false, false);
        s2 = __builtin_amdgcn_wmma_f32_16x16x4_f32(false, band[k], false, b2,
                                                   (short)0, s2, false, false);
    }

    // ---- Stage 4: fused epilogue (saturation * contrast / exposedness) ----
    const float inv = 1.0f / (25.0f * 25.0f * 3.0f);   // box mean + channel mean
    const float EPS = 1.0f / 255.0f;
    float* outb = out + (size_t)b * CH_STRIDE;
    #pragma unroll
    for (int v = 0; v < 8; ++v) {
        const int row = tr + v + 8 * hi;
        const int col = tc + lo;
        const size_t off = (size_t)row * W_IMG + col;
        float meanr = s1[v] * inv;
        float means = s2[v] * inv;
        float expo  = fabsf(meanr - 0.5f) + EPS;
        float contrast = means - meanr * meanr;
        float x0 = base[off];
        float x1 = base[off + CH_STRIDE];
        float x2 = base[off + 2 * CH_STRIDE];
        float mx = fmaxf(x0, fmaxf(x1, x2));
        float mn = fminf(x0, fminf(x1, x2));
        float sat = (mx - mn + EPS) / (mx + EPS);
        outb[off] = sat * contrast / expo;
    }
}

extern "C" void kernel_launch(void* const* d_in, const int* in_sizes, int n_in,
                              void* d_out, int out_size, void* d_ws, size_t ws_size,
                              hipStream_t stream) {
    (void)in_sizes; (void)n_in; (void)d_ws; (void)ws_size; (void)out_size;
    const float* img = (const float*)d_in[0];     // [16,3,512,512] f32
    float* out = (float*)d_out;                   // [16,1,512,512] f32
    dim3 grid(W_IMG / 16, H_IMG / 16, 16);        // 32 x 32 tiles x 16 images
    ImageEvaluator_84000970375834_kernel<<<grid, dim3(32), 0, stream>>>(img, out);
}